// SRR_11527692222498
// MI455X (gfx1250) — compile-verified
//
#include <hip/hip_runtime.h>
#include <cstdint>
#include <cstddef>

// ---------------- problem constants (match reference) ----------------
#define NITER 10
#define ALPHA_C 0.5f
// BETA = 0.02*DELTA^2 = 2e-4 ; ALPHA*BETA = 1e-4
#define ALPHABETA_C 1.0e-4f
#define INV_D2 100.0f          // 1/DELTA^2

#define Dv 192
#define Hv 192
#define Wv 192
#define NS 192
#define SH 192
#define SW 192
#define VOL (Dv * Hv * Wv)
#define NPIX (NS * SH * SW)

// stencil tile
#define TX 16
#define TY 16
#define TZ 4
#define HX (TX + 2)
#define HY (TY + 2)
#define HZ (TZ + 2)
#define HALO_ELEMS (HX * HY * HZ)

// TDM builtin arity differs between toolchains (see CDNA5_HIP.md):
// therock headers ship the TDM descriptor header and use the 6-arg form.
#if __has_include(<hip/amd_detail/amd_gfx1250_TDM.h>)
#define TDM_6ARG 1
#endif

typedef unsigned int v4u __attribute__((ext_vector_type(4)));
typedef int v4i __attribute__((ext_vector_type(4)));
typedef int v8i __attribute__((ext_vector_type(8)));

__device__ __forceinline__ int iclamp(int v, int lo, int hi) {
    return v < lo ? lo : (v > hi ? hi : v);
}

// Compute trilinear base coords + fractions for one slice pixel.
__device__ __forceinline__ void srr_coeffs(const float* __restrict__ theta,
                                           int n, int row, int col,
                                           float& x0, float& y0, float& z0,
                                           float& fx, float& fy, float& fz) {
    const float px = (float)col - 0.5f * (float)(SW - 1);
    const float py = (float)row - 0.5f * (float)(SH - 1);
    const float* th = theta + n * 12;
    // pts = (px, py, 0); pv = R*pts + t
    float ix = th[0] * px + th[1] * py + th[3]  + 0.5f * (float)(Wv - 1);
    float iy = th[4] * px + th[5] * py + th[7]  + 0.5f * (float)(Hv - 1);
    float iz = th[8] * px + th[9] * py + th[11] + 0.5f * (float)(Dv - 1);
    x0 = floorf(ix); y0 = floorf(iy); z0 = floorf(iz);
    fx = ix - x0;    fy = iy - y0;    fz = iz - z0;
}

// ---------------- kernel 1: err = A(x) - slices ----------------
// grid(SH, NS), block(SW): n/row are block-uniform (scalar theta load),
// col = threadIdx.x -> no integer division, coalesced gathers.
__global__ __launch_bounds__(SW) void srr_err_kernel(
    const float* __restrict__ theta, const float* __restrict__ x,
    const float* __restrict__ slices, float* __restrict__ err) {
    const int col = threadIdx.x;
    const int row = blockIdx.x;
    const int n   = blockIdx.y;

    float x0, y0, z0, fx, fy, fz;
    srr_coeffs(theta, n, row, col, x0, y0, z0, fx, fy, fz);

    float acc = 0.0f;
#pragma unroll
    for (int c = 0; c < 8; ++c) {   // order: cz,cy,cx (cx fastest) matches ref
        const int cx = c & 1, cy = (c >> 1) & 1, cz = (c >> 2) & 1;
        float xi = x0 + (float)cx, yi = y0 + (float)cy, zi = z0 + (float)cz;
        float w = (cx ? fx : 1.0f - fx) * (cy ? fy : 1.0f - fy) * (cz ? fz : 1.0f - fz);
        bool valid = (xi >= 0.0f) && (xi < (float)Wv) && (yi >= 0.0f) &&
                     (yi < (float)Hv) && (zi >= 0.0f) && (zi < (float)Dv);
        if (valid && (w != 0.0f)) {
            int xii = iclamp((int)xi, 0, Wv - 1);
            int yii = iclamp((int)yi, 0, Hv - 1);
            int zii = iclamp((int)zi, 0, Dv - 1);
            acc += x[((size_t)zii * Hv + yii) * Wv + xii] * w;
        }
    }
    const size_t i = ((size_t)n * SH + row) * SW + col;
    err[i] = acc - slices[i];
}

// ---------------- kernel 2: y = x - ALPHA*BETA*dR(x) ----------------
// 26-tap Huber-like TV gradient. Halo tile (18x18x6) staged with ONE
// Tensor Data Mover DMA per block (tensor_load_to_lds, TENSORcnt),
// then all 26 taps are served from LDS.
__global__ __launch_bounds__(256) void srr_stencil_kernel(
    const float* __restrict__ x, float* __restrict__ y) {
    __shared__ float tile[HALO_ELEMS];

    const int ox = blockIdx.x * TX;
    const int oy = blockIdx.y * TY;
    const int oz = blockIdx.z * TZ;
    // Clamp tile origin so the 18x18x6 box lies fully inside the volume.
    // (Halo cells outside the volume are never consumed: only interior
    // voxels run the stencil, and their neighbors are always in-bounds.)
    const int sx = iclamp(ox - 1, 0, Wv - HX);
    const int sy = iclamp(oy - 1, 0, Hv - HY);
    const int sz = iclamp(oz - 1, 0, Dv - HZ);
    const int tid = threadIdx.y * TX + threadIdx.x;

    if (tid < 32) {   // wave 0 issues the TDM descriptor (EXEC ignored by TDM)
        unsigned long long ga = (unsigned long long)(uintptr_t)(
            x + (((size_t)sz * Hv + sy) * Wv + sx));
        unsigned lds = (unsigned)(uintptr_t)(&tile[0]);   // low 32b = LDS offset

        v4u g0;
        g0.x = 1u;                                        // count=1, user D#
        g0.y = lds;                                       // lds_addr
        g0.z = (unsigned)(ga & 0xffffffffull);            // global_addr[31:0]
        g0.w = (unsigned)((ga >> 32) & 0x01ffffffull)     // global_addr[56:32]
               | 0x80000000u;                             // type=2 ("image")

        v8i g1;
        g1[0] = (int)(2u << 16);                 // data_size=4B; mask=0; no pad
        g1[1] = (int)((unsigned)Wv << 16);       // tensor_dim0[15:0] @ [63:48]
        g1[2] = (int)((unsigned)Hv << 16);       // tensor_dim1[15:0] @ [95:80]
        g1[3] = (int)((unsigned)HX << 16);       // tile_dim0 @ [127:112]
        g1[4] = (int)(((unsigned)HZ << 16) | (unsigned)HY); // tile_dim1, tile_dim2
        g1[5] = Wv;                              // tensor_dim0_stride lo32
        g1[6] = (int)(((unsigned)(Wv * Hv) & 0xffffu) << 16); // dim1_stride[15:0]
        g1[7] = (int)((unsigned)(Wv * Hv) >> 16);             // dim1_stride[47:16]

        v4i g2;
        g2[0] = Dv;                              // tensor_dim2
        g2[1] = 0;                               // tensor_dim3 / lds_incr
        g2[2] = 0;                               // tensor_dim2_stride lo (unused, tile_dim3=0)
        g2[3] = 0;                               // tile_dim3=0
        v4i g3 = {0, 0, 0, 0};

#ifdef TDM_6ARG
        v8i gz8 = {0, 0, 0, 0, 0, 0, 0, 0};
        __builtin_amdgcn_tensor_load_to_lds(g0, g1, g2, g3, gz8, 0);
#else
        __builtin_amdgcn_tensor_load_to_lds(g0, g1, g2, g3, 0);
#endif
        __builtin_amdgcn_s_wait_tensorcnt(0);
    }
    __syncthreads();

#pragma unroll
    for (int zz = 0; zz < TZ; ++zz) {
        const int gx = ox + threadIdx.x;
        const int gy = oy + threadIdx.y;
        const int gz = oz + zz;
        const int lx = gx - sx;
        const int ly = gy - sy;
        const int lz = gz - sz;
        float core = tile[(lz * HY + ly) * HX + lx];
        float out = core;
        if (gx > 0 && gx < Wv - 1 && gy > 0 && gy < Hv - 1 && gz > 0 && gz < Dv - 1) {
            float acc = 0.0f;
#pragma unroll
            for (int dx = -1; dx <= 1; ++dx)
#pragma unroll
                for (int dy = -1; dy <= 1; ++dy)
#pragma unroll
                    for (int dz = -1; dz <= 1; ++dz) {
                        if (dx == 0 && dy == 0 && dz == 0) continue;
                        float nb = tile[((lz + dz) * HY + (ly + dy)) * HX + (lx + dx)];
                        float dv = core - nb;
                        const float k = INV_D2 / (float)(dx * dx + dy * dy + dz * dz);
                        float dvs = dv * k;
                        acc += dvs * __builtin_amdgcn_rsqf(1.0f + dv * dvs);
                    }
            out = core - ALPHABETA_C * acc;
        }
        y[(((size_t)gz) * Hv + gy) * Wv + gx] = out;
    }
}

// ---------------- kernel 3: y -= ALPHA * At(err) (scatter atomics) ----------------
__global__ __launch_bounds__(SW) void srr_scatter_kernel(
    const float* __restrict__ theta, const float* __restrict__ err,
    float* __restrict__ y) {
    const int col = threadIdx.x;
    const int row = blockIdx.x;
    const int n   = blockIdx.y;

    float x0, y0, z0, fx, fy, fz;
    srr_coeffs(theta, n, row, col, x0, y0, z0, fx, fy, fz);

    const float e = err[((size_t)n * SH + row) * SW + col] * (-ALPHA_C);
#pragma unroll
    for (int c = 0; c < 8; ++c) {
        const int cx = c & 1, cy = (c >> 1) & 1, cz = (c >> 2) & 1;
        float xi = x0 + (float)cx, yi = y0 + (float)cy, zi = z0 + (float)cz;
        float w = (cx ? fx : 1.0f - fx) * (cy ? fy : 1.0f - fy) * (cz ? fz : 1.0f - fz);
        bool valid = (xi >= 0.0f) && (xi < (float)Wv) && (yi >= 0.0f) &&
                     (yi < (float)Hv) && (zi >= 0.0f) && (zi < (float)Dv);
        if (valid && (w != 0.0f)) {
            int xii = iclamp((int)xi, 0, Wv - 1);
            int yii = iclamp((int)yi, 0, Hv - 1);
            int zii = iclamp((int)zi, 0, Dv - 1);
            atomicAdd(&y[((size_t)zii * Hv + yii) * Wv + xii], e * w);
        }
    }
}

// ---------------- kernel 4: out = relu(x) ----------------
__global__ __launch_bounds__(256) void srr_relu_kernel(
    const float* __restrict__ x, float* __restrict__ out) {
    int i = blockIdx.x * blockDim.x + threadIdx.x;
    if (i < VOL) out[i] = fmaxf(x[i], 0.0f);
}

// ---------------- host orchestration ----------------
extern "C" void kernel_launch(void* const* d_in, const int* in_sizes, int n_in,
                              void* d_out, int out_size, void* d_ws, size_t ws_size,
                              hipStream_t stream) {
    (void)in_sizes; (void)n_in; (void)out_size; (void)ws_size;
    const float* theta  = (const float*)d_in[0];
    const float* slices = (const float*)d_in[1];
    const float* volume = (const float*)d_in[2];
    float* out = (float*)d_out;

    float* xbuf = (float*)d_ws;
    float* ybuf = xbuf + VOL;
    float* ebuf = ybuf + VOL;

    hipMemcpyAsync(xbuf, volume, (size_t)VOL * sizeof(float),
                   hipMemcpyDeviceToDevice, stream);

    const dim3 pgrid(SH, NS);          // (row, slice)
    const dim3 pblock(SW, 1, 1);       // col
    const dim3 sgrid(Wv / TX, Hv / TY, Dv / TZ);
    const dim3 sblock(TX, TY, 1);

    for (int it = 0; it < NITER; ++it) {
        srr_err_kernel<<<pgrid, pblock, 0, stream>>>(theta, xbuf, slices, ebuf);
        srr_stencil_kernel<<<sgrid, sblock, 0, stream>>>(xbuf, ybuf);
        srr_scatter_kernel<<<pgrid, pblock, 0, stream>>>(theta, ebuf, ybuf);
        float* t = xbuf; xbuf = ybuf; ybuf = t;   // fixed swap sequence: deterministic
    }
    srr_relu_kernel<<<(VOL + 255) / 256, 256, 0, stream>>>(xbuf, out);
}